// LightSHStepMLP_9594956939354
// MI455X (gfx1250) — compile-verified
//
#include <hip/hip_runtime.h>
#include <hip/hip_bf16.h>

typedef __attribute__((ext_vector_type(2))) float v2f;
typedef __attribute__((ext_vector_type(4))) float f4;
typedef __attribute__((ext_vector_type(8))) float v8f;

#define SQRT_PI_F        1.7724538509055159f
#define SQRT_PI_OVER_3_F 1.0233267079464885f
#define K1_F             0.4886025119029199f
#define HALF_SQRT_PI_F   0.8862269254527580f
#define INV_SQRT_PI_F    0.5641895835477563f

// -------------------------------------------------------------------------
// Stage A: build the 256-entry material table with V_WMMA_F32_16X16X4_F32.
// Grid: 16 blocks x 32 threads (1 wave per block, EXEC all-ones for WMMA).
// Block b handles materials [16b, 16b+16).
//   H(16x24)  = relu(EMB(16x8) @ FC1^T(8x24) + b1)     : 2 N-tiles, 2 K-steps
//   OUT(16x6) = H(16x24) @ [alb|emit]^T(24x6) + bias2  : 6 K-steps
// table[m*8 + c]: c=0..2 albedo, c=3..5 emit (stride 8 for alignment).
// -------------------------------------------------------------------------
__global__ __launch_bounds__(32) void mlp_table_kernel(
    const float* __restrict__ emb,
    const float* __restrict__ fc1_w, const float* __restrict__ fc1_b,
    const float* __restrict__ alb_w, const float* __restrict__ alb_b,
    const float* __restrict__ emit_w, const float* __restrict__ emit_b,
    float* __restrict__ table)
{
    __shared__ float lds_h[16 * 24];

    const int lane = threadIdx.x;     // 0..31
    const int lm   = lane & 15;       // M (A-matrix), N (B/C/D matrices)
    const int lh   = lane >> 4;       // half-wave selector
    const int m0   = blockIdx.x * 16; // material row block

    // ---- Stage 1: H = relu(EMB @ FC1^T + b1) -----------------------------
    #pragma unroll
    for (int nt = 0; nt < 2; ++nt) {          // hidden cols [16*nt, 16*nt+16)
        v8f acc = {};
        const int j = nt * 16 + lm;           // hidden unit for B / bias
        #pragma unroll
        for (int k0 = 0; k0 < 8; k0 += 4) {   // K = D_EMB = 8
            const int ka = k0 + 2 * lh;
            v2f a, b;
            // A: 16x4 f32 tile of EMB (row-major 256x8)
            a.x = emb[(m0 + lm) * 8 + ka + 0];
            a.y = emb[(m0 + lm) * 8 + ka + 1];
            // B: 4x16 f32 tile of FC1^T; B[k][j] = fc1_w[j][k] (fc1_w: 24x8)
            if (j < 24) { b.x = fc1_w[j * 8 + ka + 0]; b.y = fc1_w[j * 8 + ka + 1]; }
            else        { b.x = 0.0f;                  b.y = 0.0f;                  }
            acc = __builtin_amdgcn_wmma_f32_16x16x4_f32(
                false, a, false, b, (short)0, acc, false, false);
        }
        if (j < 24) {
            const float bias = fc1_b[j];
            #pragma unroll
            for (int r = 0; r < 8; ++r) {     // D: VGPR r -> row (r + 8*lh)
                float h = acc[r] + bias;
                h = h > 0.0f ? h : 0.0f;
                lds_h[(r + 8 * lh) * 24 + j] = h;
            }
        }
    }
    __syncthreads();  // cross-lane LDS dependency (s_wait_dscnt; barrier=NOP, 1 wave)

    // ---- Stage 2: OUT = H @ W2 + bias2 (cols 0-2 albedo, 3-5 emit) -------
    float bias2 = 0.0f;
    if (lm < 3)      bias2 = alb_b[lm];
    else if (lm < 6) bias2 = emit_b[lm - 3];
    v8f acc2 = { bias2, bias2, bias2, bias2, bias2, bias2, bias2, bias2 };

    #pragma unroll
    for (int k0 = 0; k0 < 24; k0 += 4) {      // K = D_HID = 24
        const int ka = k0 + 2 * lh;
        v2f a, b;
        a.x = lds_h[lm * 24 + ka + 0];        // A: H tile (row = lm)
        a.y = lds_h[lm * 24 + ka + 1];
        // B[k][c]: c<3 -> alb_w[c][k], c<6 -> emit_w[c-3][k], else 0
        if (lm < 3)      { b.x = alb_w[lm * 24 + ka];        b.y = alb_w[lm * 24 + ka + 1]; }
        else if (lm < 6) { b.x = emit_w[(lm - 3) * 24 + ka]; b.y = emit_w[(lm - 3) * 24 + ka + 1]; }
        else             { b.x = 0.0f;                       b.y = 0.0f; }
        acc2 = __builtin_amdgcn_wmma_f32_16x16x4_f32(
            false, a, false, b, (short)0, acc2, false, false);
    }
    if (lm < 6) {
        #pragma unroll
        for (int r = 0; r < 8; ++r)
            table[(m0 + r + 8 * lh) * 8 + lm] = acc2[r];
    }
}

// -------------------------------------------------------------------------
// Stage B: HBM-bound streaming pass. 112 B moved per point -> ~19 us at
// 23.3 TB/s. float4 (B128) loads/stores with NT hints; 8 KB table stays hot.
// -------------------------------------------------------------------------
__global__ __launch_bounds__(256) void shade_kernel(
    const int*   __restrict__ material,
    const float* __restrict__ normal,
    const float* __restrict__ sh_in,
    const float* __restrict__ table,
    float*       __restrict__ out, int n)
{
    const int i = blockIdx.x * 256 + threadIdx.x;
    if (i >= n) return;

    const int mat = material[i];

    const f4* shp = (const f4*)(sh_in + (size_t)i * 12);
    const f4 s0 = __builtin_nontemporal_load(shp + 0); // c0.rgb, cy.r
    const f4 s1 = __builtin_nontemporal_load(shp + 1); // cy.gb, cz.rg
    const f4 s2 = __builtin_nontemporal_load(shp + 2); // cz.b, cx.rgb

    const float* np = normal + (size_t)i * 3;
    const float nx = np[0], ny = np[1], nz = np[2];

    const float c0[3] = { s0.x, s0.y, s0.z };
    const float cy[3] = { s0.w, s1.x, s1.y };
    const float cz[3] = { s1.z, s1.w, s2.x };
    const float cx[3] = { s2.y, s2.z, s2.w };

    float alb[3], emt[3];
    #pragma unroll
    for (int c = 0; c < 3; ++c) {
        alb[c] = table[mat * 8 + c];
        emt[c] = table[mat * 8 + 3 + c];
    }

    const bool isEmpty = (mat == 0);
    const bool isLight = (mat == 1);
    const float k1ny = K1_F * ny, k1nz = K1_F * nz, k1nx = K1_F * nx;

    float o[12];
    #pragma unroll
    for (int c = 0; c < 3; ++c) {
        const float E      = HALF_SQRT_PI_F * c0[c]
                           - SQRT_PI_OVER_3_F * (cx[c] * nx + cy[c] * ny + cz[c] * nz);
        const float common = alb[c] * E;
        const float so0 = common * INV_SQRT_PI_F;
        const float so1 = common * k1ny;
        const float so2 = common * k1nz;
        const float so3 = common * k1nx;
        const float lo0 =  emt[c] * HALF_SQRT_PI_F;
        const float lo1 = -emt[c] * SQRT_PI_OVER_3_F;
        o[0 + c] = isEmpty ? c0[c] : (isLight ? lo0 : so0);
        o[3 + c] = isEmpty ? cy[c] : (isLight ? lo1 : so1);
        o[6 + c] = isEmpty ? cz[c] : (isLight ? 0.0f : so2);
        o[9 + c] = isEmpty ? cx[c] : (isLight ? 0.0f : so3);
    }

    f4* op = (f4*)(out + (size_t)i * 12);
    const f4 w0 = { o[0], o[1], o[2],  o[3] };
    const f4 w1 = { o[4], o[5], o[6],  o[7] };
    const f4 w2 = { o[8], o[9], o[10], o[11] };
    __builtin_nontemporal_store(w0, op + 0);
    __builtin_nontemporal_store(w1, op + 1);
    __builtin_nontemporal_store(w2, op + 2);
}

extern "C" void kernel_launch(void* const* d_in, const int* in_sizes, int n_in,
                              void* d_out, int out_size, void* d_ws, size_t ws_size,
                              hipStream_t stream) {
    const int*   material = (const int*)d_in[0];
    const float* normal   = (const float*)d_in[1];
    const float* sh_in    = (const float*)d_in[2];
    const float* emb      = (const float*)d_in[3];
    const float* fc1_w    = (const float*)d_in[4];
    const float* fc1_b    = (const float*)d_in[5];
    const float* alb_w    = (const float*)d_in[6];
    const float* alb_b    = (const float*)d_in[7];
    const float* emit_w   = (const float*)d_in[8];
    const float* emit_b   = (const float*)d_in[9];
    float* out   = (float*)d_out;
    float* table = (float*)d_ws;   // 256 * 8 floats = 8 KB scratch

    const int n = in_sizes[0];

    // Stage A: 256-material MLP table via f32 WMMA (16 tiles of 16 materials).
    mlp_table_kernel<<<16, 32, 0, stream>>>(emb, fc1_w, fc1_b,
                                            alb_w, alb_b, emit_w, emit_b, table);

    // Stage B: bandwidth-bound shading pass.
    const int blocks = (n + 255) / 256;
    shade_kernel<<<blocks, 256, 0, stream>>>(material, normal, sh_in, table, out, n);
}